// CubicalLayer_1614907703765
// MI455X (gfx1250) — compile-verified
//
#include <hip/hip_runtime.h>
#include <stdint.h>

#define RED_BLOCK 256
#define RED_GRID  1024
#define GAT_BLOCK 256

typedef float v4f __attribute__((ext_vector_type(4)));
typedef int   v4i __attribute__((ext_vector_type(4)));

// Monotonic float->uint key: min(float) <-> min(uint key).
__device__ __forceinline__ uint32_t f32_key(float f) {
    uint32_t u = __float_as_uint(f);
    return (u & 0x80000000u) ? ~u : (u | 0x80000000u);
}
__device__ __forceinline__ float key_f32(uint32_t k) {
    uint32_t u = (k & 0x80000000u) ? (k & 0x7fffffffu) : ~k;
    return __uint_as_float(u);
}

// Streaming min-reduction over X using the CDNA5 async global->LDS path,
// double-buffered, tracked with ASYNCcnt. Each lane copies its own 16B slot,
// so the per-wave s_wait_asynccnt suffices (no cross-wave sharing, no barrier).
// Each block writes its min KEY to ws[blockIdx] -> no init kernel, no atomics,
// deterministic by construction.
__global__ __launch_bounds__(RED_BLOCK) void reduce_min_async_kernel(
    const float* __restrict__ X, unsigned n4, uint32_t* __restrict__ ws) {
    __shared__ v4f tile[2][RED_BLOCK];
    __shared__ uint32_t wavemin[RED_BLOCK / 32];

    const unsigned tid = threadIdx.x;
    const unsigned gsz = RED_GRID * RED_BLOCK;
    const unsigned gid = blockIdx.x * RED_BLOCK + tid;
    const unsigned iters = n4 / gsz;  // uniform across all waves
    const v4f* __restrict__ X4 = (const v4f*)X;

    const uint32_t lds0 = (uint32_t)(uintptr_t)&tile[0][tid];
    const uint32_t lds1 = (uint32_t)(uintptr_t)&tile[1][tid];

    float m = __builtin_inff();

    if (iters) {
        unsigned i = gid;
        asm volatile("global_load_async_to_lds_b128 %0, %1, off"
                     :: "v"(lds0),
                        "v"((unsigned long long)(uintptr_t)(X4 + i))
                     : "memory");
        for (unsigned k = 0; k < iters; ++k) {
            const unsigned buf = k & 1u;
            if (k + 1 < iters) {  // uniform branch
                asm volatile("global_load_async_to_lds_b128 %0, %1, off"
                             :: "v"(buf ? lds0 : lds1),
                                "v"((unsigned long long)(uintptr_t)(X4 + i + gsz))
                             : "memory");
                asm volatile("s_wait_asynccnt 0x1" ::: "memory");
            } else {
                asm volatile("s_wait_asynccnt 0x0" ::: "memory");
            }
            v4f v = tile[buf][tid];
            m = fminf(m, fminf(fminf(v.x, v.y), fminf(v.z, v.w)));
            i += gsz;
        }
    }
    // Generic tail (empty for 4096x4096).
    for (unsigned i = iters * gsz + gid; i < n4; i += gsz) {
        v4f v = X4[i];
        m = fminf(m, fminf(fminf(v.x, v.y), fminf(v.z, v.w)));
    }

    // wave32 tree reduction
    for (int off = 16; off > 0; off >>= 1)
        m = fminf(m, __shfl_xor(m, off, 32));
    if ((tid & 31u) == 0) wavemin[tid >> 5] = f32_key(m);
    __syncthreads();
    if (tid == 0) {
        uint32_t k2 = wavemin[0];
#pragma unroll
        for (int j = 1; j < RED_BLOCK / 32; ++j) k2 = min(k2, wavemin[j]);
        ws[blockIdx.x] = k2;
    }
}

// Single block reduces RED_GRID per-block keys and writes the essential value.
__global__ __launch_bounds__(RED_BLOCK) void finalize_min_kernel(
    const uint32_t* __restrict__ ws, float* __restrict__ out_essential) {
    __shared__ uint32_t wavemin[RED_BLOCK / 32];
    const unsigned tid = threadIdx.x;
    uint32_t k = 0xFFFFFFFFu;
#pragma unroll
    for (int j = 0; j < RED_GRID / RED_BLOCK; ++j)
        k = min(k, ws[tid + j * RED_BLOCK]);
    for (int off = 16; off > 0; off >>= 1)
        k = min(k, (uint32_t)__shfl_xor((int)k, off, 32));
    if ((tid & 31u) == 0) wavemin[tid >> 5] = k;
    __syncthreads();
    if (tid == 0) {
        uint32_t k2 = wavemin[0];
#pragma unroll
        for (int j = 1; j < RED_BLOCK / 32; ++j) k2 = min(k2, wavemin[j]);
        out_essential[0] = key_f32(k2);
    }
}

// One launch covering BOTH gathers: blocks [0,g0) -> idx0/out0, [g0,g0+g1) ->
// idx1/out1. 4 elems/thread: NT vector index loads + prefetch (stream once,
// keep L2 for X), regular-temporal X loads (L2-resident after the reduction),
// NT scalar stores (out1 is only 4B-aligned; scalar b32 keeps both segments
// uniform and fully coalesced).
__global__ __launch_bounds__(GAT_BLOCK) void gather_both_kernel(
    const float* __restrict__ X,
    const int* __restrict__ idx0, float* __restrict__ out0, unsigned q0, unsigned g0,
    const int* __restrict__ idx1, float* __restrict__ out1, unsigned q1) {
    const int* __restrict__ idx;
    float* __restrict__ out;
    unsigned t, q;
    if (blockIdx.x < g0) {
        idx = idx0; out = out0; q = q0;
        t = blockIdx.x * GAT_BLOCK + threadIdx.x;
    } else {
        idx = idx1; out = out1; q = q1;
        t = (blockIdx.x - g0) * GAT_BLOCK + threadIdx.x;
    }
    if (t >= q) return;
    const v4i* p = (const v4i*)idx;
    if (t + 4096 < q)
        __builtin_prefetch((const void*)(p + t + 4096), 0, 0);
    v4i v = __builtin_nontemporal_load(p + t);
    float a = X[v.x];
    float b = X[v.y];
    float c = X[v.z];
    float d = X[v.w];
    float* o = out + (size_t)t * 4;
    __builtin_nontemporal_store(a, o + 0);
    __builtin_nontemporal_store(b, o + 1);
    __builtin_nontemporal_store(c, o + 2);
    __builtin_nontemporal_store(d, o + 3);
}

// Scalar remainder gather (not launched for the 4M-index case).
__global__ void gather1_kernel(const float* __restrict__ X,
                               const int* __restrict__ idx,
                               float* __restrict__ out, unsigned n) {
    unsigned t = blockIdx.x * blockDim.x + threadIdx.x;
    if (t < n) __builtin_nontemporal_store(X[idx[t]], out + t);
}

extern "C" void kernel_launch(void* const* d_in, const int* in_sizes, int n_in,
                              void* d_out, int out_size, void* d_ws, size_t ws_size,
                              hipStream_t stream) {
    const float* X   = (const float*)d_in[0];
    const int* idx0  = (const int*)d_in[1];
    const int* idx1  = (const int*)d_in[2];
    float* out       = (float*)d_out;
    uint32_t* ws     = (uint32_t*)d_ws;  // needs RED_GRID * 4 bytes

    const unsigned nX = (unsigned)in_sizes[0];  // 16,777,216
    const unsigned n0 = (unsigned)in_sizes[1];  // 4,000,000
    const unsigned n1 = (unsigned)in_sizes[2];  // 4,000,000

    // 1) min-reduction first: streams X once (async->LDS), warming the 192MB
    //    L2 so the random gathers below hit L2 instead of HBM.
    reduce_min_async_kernel<<<RED_GRID, RED_BLOCK, 0, stream>>>(X, nX / 4, ws);
    finalize_min_kernel<<<1, RED_BLOCK, 0, stream>>>(ws, out + n0);  // essential0

    // 2) both gathers in one launch
    float* out1 = out + n0 + 1;  // finite1 starts after essential0
    const unsigned q0 = n0 / 4, q1 = n1 / 4;
    const unsigned g0 = (q0 + GAT_BLOCK - 1) / GAT_BLOCK;
    const unsigned g1 = (q1 + GAT_BLOCK - 1) / GAT_BLOCK;
    if (g0 + g1)
        gather_both_kernel<<<g0 + g1, GAT_BLOCK, 0, stream>>>(
            X, idx0, out, q0, g0, idx1, out1, q1);

    // Remainders (empty when n0/n1 are multiples of 4, as here).
    if (n0 & 3u)
        gather1_kernel<<<1, 256, 0, stream>>>(X, idx0 + q0 * 4, out + (size_t)q0 * 4, n0 & 3u);
    if (n1 & 3u)
        gather1_kernel<<<1, 256, 0, stream>>>(X, idx1 + q1 * 4, out1 + (size_t)q1 * 4, n1 & 3u);
    // essential1 is a 0-element tensor: nothing to write.
}